// LETTERRQBottleneck_71923522339243
// MI455X (gfx1250) — compile-verified
//
#include <hip/hip_runtime.h>
#include <hip/hip_bf16.h>

// Residual quantization (RQ bottleneck) for MI455X / gfx1250.
// N=65536 rows, D=512, K=1024 codes, L=4 levels, fp32.
//
// argmin_k(||r||^2+||c_k||^2-2 r.c_k) == argmin_k(0.5||c_k||^2 - r.c_k), so the
// hot loop is just R @ C^T on the f32 WMMA pipe plus a precomputed 0.5||c||^2
// table. quant = input - final_residual.
//
// v3: unroll-8 K loop (32 B-loads in flight per clause) + global_prefetch_b8
// warming of the next N-pass's codebook lines to shrink the L2-latency bubble
// at pass boundaries (occupancy is LDS-limited to 2 WGs/WGP -> 2 waves/SIMD).

#define RQ_N 65536
#define RQ_D 512
#define RQ_K 1024
#define RQ_L 4
#define ROWS_PER_WG 64   // 4 waves x 16 rows

typedef __attribute__((ext_vector_type(2))) float v2f;
typedef __attribute__((ext_vector_type(8))) float v8f;

// ---------------------------------------------------------------------------
// Kernel 1: csq[l*K + k] = 0.5 * sum_d cb[l][k][d]^2   (one wave per (l,k) row)
// ---------------------------------------------------------------------------
__global__ __launch_bounds__(256) void rq_csq_kernel(const float* __restrict__ cb,
                                                     float* __restrict__ csq) {
    const int row  = blockIdx.x * 8 + (threadIdx.x >> 5);   // 0 .. L*K-1
    const int lane = threadIdx.x & 31;
    const float* p = cb + (size_t)row * RQ_D;
    float s = 0.0f;
    #pragma unroll
    for (int i = 0; i < RQ_D / 32; ++i) {
        float v = p[lane + 32 * i];
        s += v * v;
    }
    #pragma unroll
    for (int off = 16; off >= 1; off >>= 1)
        s += __shfl_xor(s, off, 32);
    if (lane == 0) csq[row] = 0.5f * s;
}

// ---------------------------------------------------------------------------
// Kernel 2: main residual-quantization kernel.
// 1 WG = 128 threads (4 waves), owns 64 rows. Residual kept in LDS.
// Wave w computes the 16-row stripe [w*16, w*16+16) with f32 WMMA,
// 4 N-tiles (64 codes) per pass with independent accumulator chains.
// ---------------------------------------------------------------------------
__global__ __launch_bounds__(128) void rq_main_kernel(const float* __restrict__ in,
                                                      const float* __restrict__ cb,
                                                      float* __restrict__ outQ,
                                                      int* __restrict__ outC,
                                                      const float* __restrict__ csq) {
    __shared__ float sh_res[ROWS_PER_WG * RQ_D];   // 128 KB residual tile
    __shared__ float sh_csq[RQ_K];                 // 4 KB half-squared norms
    __shared__ int   sh_idx[ROWS_PER_WG];          // winning code per row

    const int tid     = threadIdx.x;
    const int wave    = tid >> 5;
    const int lane    = tid & 31;
    const int half    = lane >> 4;          // 0: lanes 0-15, 1: lanes 16-31
    const int l15     = lane & 15;
    const int m0      = wave * 16;          // wave's row stripe within WG tile
    const int rowBase = blockIdx.x * ROWS_PER_WG;

    // ---- load this block's input rows into LDS residual (coalesced) ----
    const size_t gbase = (size_t)rowBase * RQ_D;
    for (int i = tid; i < ROWS_PER_WG * RQ_D; i += 128)
        sh_res[i] = in[gbase + i];
    __syncthreads();

    // A-fragment base for V_WMMA_F32_16X16X4_F32 (16x4, MxK):
    //   lanes 0-15: M=lane,   VGPR0=K+0, VGPR1=K+1
    //   lanes16-31: M=lane-16,VGPR0=K+2, VGPR1=K+3
    const float* aRow = &sh_res[(m0 + l15) * RQ_D + 2 * half];

    for (int lvl = 0; lvl < RQ_L; ++lvl) {
        const float* cbL = cb + (size_t)lvl * RQ_K * RQ_D;
        const bool lastLvl = (lvl == RQ_L - 1);

        // stage this level's 0.5*||c||^2 table into LDS
        for (int i = tid; i < RQ_K; i += 128)
            sh_csq[i] = csq[lvl * RQ_K + i];
        __syncthreads();

        float bestv[8];
        int   besti[8];
        #pragma unroll
        for (int j = 0; j < 8; ++j) { bestv[j] = 3.4e38f; besti[j] = 0; }

        // ---- score GEMM: 16 passes of 4 N-tiles, K-reduce 512 by 4 ----
        for (int n0 = 0; n0 < RQ_K; n0 += 64) {
            const int n = n0 + l15;
            // B-fragment (4x16, KxN), mirror of A layout:
            //   lanes 0-15: N=lane,    K offsets 0,1
            //   lanes16-31: N=lane-16, K offsets 2,3
            const float* bRow = cbL + (size_t)n * RQ_D + 2 * half;
            const bool doPrefetch = (n0 + 64 < RQ_K) || !lastLvl;

            v8f acc0 = {}, acc1 = {}, acc2 = {}, acc3 = {};
            #pragma unroll 8
            for (int k = 0; k < RQ_D; k += 4) {
                // Warm the next N-pass's codebook lines one 128B line at a
                // time (each line covers 8 k-steps for this lane).
                if ((k & 31) == 0 && doPrefetch) {
                    __builtin_prefetch(bRow + 64 * RQ_D + k, 0, 3);
                    __builtin_prefetch(bRow + 80 * RQ_D + k, 0, 3);
                    __builtin_prefetch(bRow + 96 * RQ_D + k, 0, 3);
                    __builtin_prefetch(bRow + 112 * RQ_D + k, 0, 3);
                }
                v2f a  = *(const v2f*)(aRow + k);
                v2f b0 = *(const v2f*)(bRow + k);
                v2f b1 = *(const v2f*)(bRow + 16 * RQ_D + k);
                v2f b2 = *(const v2f*)(bRow + 32 * RQ_D + k);
                v2f b3 = *(const v2f*)(bRow + 48 * RQ_D + k);
                acc0 = __builtin_amdgcn_wmma_f32_16x16x4_f32(
                    false, a, false, b0, (short)0, acc0, false, false);
                acc1 = __builtin_amdgcn_wmma_f32_16x16x4_f32(
                    false, a, false, b1, (short)0, acc1, false, false);
                acc2 = __builtin_amdgcn_wmma_f32_16x16x4_f32(
                    false, a, false, b2, (short)0, acc2, false, false);
                acc3 = __builtin_amdgcn_wmma_f32_16x16x4_f32(
                    false, a, false, b3, (short)0, acc3, false, false);
            }

            // dist/2 (up to a per-row constant) = csq - dot; running argmin.
            // Tiles processed in increasing n order with strict < to keep
            // lowest-index tie-breaking within a lane.
            const float cs0 = sh_csq[n];
            const float cs1 = sh_csq[n + 16];
            const float cs2 = sh_csq[n + 32];
            const float cs3 = sh_csq[n + 48];
            #pragma unroll
            for (int j = 0; j < 8; ++j) {
                float v0 = cs0 - acc0[j];
                if (v0 < bestv[j]) { bestv[j] = v0; besti[j] = n; }
                float v1 = cs1 - acc1[j];
                if (v1 < bestv[j]) { bestv[j] = v1; besti[j] = n + 16; }
                float v2 = cs2 - acc2[j];
                if (v2 < bestv[j]) { bestv[j] = v2; besti[j] = n + 32; }
                float v3 = cs3 - acc3[j];
                if (v3 < bestv[j]) { bestv[j] = v3; besti[j] = n + 48; }
            }
        }

        // ---- cross-lane argmin over the 16 lanes holding the same row ----
        #pragma unroll
        for (int j = 0; j < 8; ++j) {
            float bv = bestv[j];
            int   bi = besti[j];
            #pragma unroll
            for (int off = 1; off < 16; off <<= 1) {
                float ov = __shfl_xor(bv, off, 32);
                int   oi = __shfl_xor(bi, off, 32);
                if (ov < bv || (ov == bv && oi < bi)) { bv = ov; bi = oi; }
            }
            bestv[j] = bv; besti[j] = bi;
        }

        if (l15 == 0) {
            #pragma unroll
            for (int j = 0; j < 8; ++j) {
                const int m = j + 8 * half;          // row within wave stripe
                const int r = m0 + m;                // row within WG tile
                sh_idx[r] = besti[j];
                outC[(size_t)(rowBase + r) * RQ_L + lvl] = besti[j];
            }
        }
        __syncthreads();

        // ---- residual update: res[r][:] -= cb[idx[r]][:] (coalesced) ----
        for (int r = 0; r < ROWS_PER_WG; ++r) {
            const float* e = cbL + (size_t)sh_idx[r] * RQ_D;
            #pragma unroll
            for (int d = 0; d < RQ_D / 128; ++d)
                sh_res[r * RQ_D + tid + d * 128] -= e[tid + d * 128];
        }
        __syncthreads();
    }

    // ---- quant = input - final residual (coalesced) ----
    for (int i = tid; i < ROWS_PER_WG * RQ_D; i += 128)
        outQ[gbase + i] = in[gbase + i] - sh_res[i];
}

extern "C" void kernel_launch(void* const* d_in, const int* in_sizes, int n_in,
                              void* d_out, int out_size, void* d_ws, size_t ws_size,
                              hipStream_t stream) {
    (void)in_sizes; (void)n_in; (void)out_size; (void)ws_size;
    const float* in = (const float*)d_in[0];                 // [N, D] f32
    const float* cb = (const float*)d_in[1];                 // [L, K, D] f32
    float* outQ = (float*)d_out;                             // [N, D] f32
    int*   outC = (int*)((float*)d_out + (size_t)RQ_N * RQ_D); // [N, L] i32 bits
    float* csq  = (float*)d_ws;                              // [L*K] f32 scratch

    // 1) half-squared norms of all codebook rows: L*K = 4096 waves
    rq_csq_kernel<<<dim3((RQ_L * RQ_K) / 8), dim3(256), 0, stream>>>(cb, csq);
    // 2) main RQ kernel: 65536/64 = 1024 workgroups x 128 threads
    rq_main_kernel<<<dim3(RQ_N / ROWS_PER_WG), dim3(128), 0, stream>>>(in, cb, outQ, outC, csq);
}